// TripletMarginLoss_1821066134162
// MI455X (gfx1250) — compile-verified
//
#include <hip/hip_runtime.h>
#include <hip/hip_bf16.h>

typedef float v2f __attribute__((ext_vector_type(2)));
typedef float v8f __attribute__((ext_vector_type(8)));

#define NB 16      // batch
#define NCH 48     // channels
#define NH 64
#define NW 64
#define NHW 4096   // H*W
#define NCORR 2048

// ---------------------------------------------------------------------------
// Kernel 1: per-pixel L2 normalization; write descriptors as [b][pixel][48]
// (contiguous 48 floats per pixel -> WMMA fragment loads are 8B vector loads)
// grid: (NB*NHW/256, 2), block 256.  blockIdx.y selects image-1 vs image-2.
// ---------------------------------------------------------------------------
__global__ __launch_bounds__(256) void k_normalize(
    const float* __restrict__ x1, const float* __restrict__ x2,
    float* __restrict__ desc1, float* __restrict__ desc2) {
  int g = blockIdx.x * 256 + threadIdx.x;          // 0 .. NB*NHW-1
  const float* x = (blockIdx.y == 0) ? x1 : x2;
  float* d       = (blockIdx.y == 0) ? desc1 : desc2;
  int b = g >> 12;
  int p = g & (NHW - 1);
  const float* src = x + (size_t)b * NCH * NHW + p;   // channel stride = NHW
  float v[NCH];
  float ss = 0.f;
#pragma unroll
  for (int c = 0; c < NCH; ++c) { v[c] = src[(size_t)c * NHW]; ss += v[c] * v[c]; }
  float inv = 1.0f / fmaxf(sqrtf(ss), 1e-12f);
  float4* dst = (float4*)(d + (size_t)g * NCH);       // 192B aligned -> float4 ok
#pragma unroll
  for (int c = 0; c < NCH / 4; ++c)
    dst[c] = make_float4(v[4 * c] * inv, v[4 * c + 1] * inv,
                         v[4 * c + 2] * inv, v[4 * c + 3] * inv);
}

// ---------------------------------------------------------------------------
// Kernel 2: fused GEMM + masked hard-negative search, in dot space.
//   Each wave owns a 16-row tile and sweeps all 4096 columns in 16-col tiles:
//   12 chained V_WMMA_F32_16X16X4_F32 (K=48), then per row:
//     masked  <=>  max_u32(riB-ci, rjB-cj) <= 8   (positions pre-biased by +4)
//     score   = masked ? -5.0 : dot     (penalty can never win the min)
//     rmax    = max(rmax, score);  final neg = 2 - 2*rmax
//   No LDS, no syncthreads; row mask positions broadcast via __shfl.
// grid: (NCORR/64, NB), block 128 = 4 waves, one 16-row tile per wave.
// pass 0 -> neg2 (rows=ids, mask=fmap_pos2); pass 1 -> neg1 (rows=p2, mask=ids)
// ---------------------------------------------------------------------------
__global__ __launch_bounds__(128) void k_negmin(
    const float* __restrict__ rowDesc, const float* __restrict__ colDesc,
    const int* __restrict__ ids, const int* __restrict__ fp2,
    float* __restrict__ out, int pass) {
  int b = blockIdx.y;
  int tid = threadIdx.x;
  int lane = tid & 31;
  int wv = tid >> 5;              // wave 0..3
  int lo = lane & 15;             // N / M-low index
  int hi = lane >> 4;             // half-select
  int rowBase = blockIdx.x * 64 + wv * 16;

  const int* idsB = ids + (size_t)b * NCORR;
  const int* fpI  = fp2 + (size_t)b * 2 * NCORR;
  const int* fpJ  = fpI + NCORR;

  // row pixel + biased mask position for this lane's row (lo)
  int n = rowBase + lo;
  int rowPix, mi, mj;
  if (pass == 0) {                // neg2: rows gathered by ids, mask from fp2
    rowPix = idsB[n];
    mi = fpI[n]; mj = fpJ[n];
  } else {                        // neg1: rows at p2, mask from ids positions
    int i2 = fpI[n], j2 = fpJ[n];
    rowPix = i2 * NW + j2;
    int id = idsB[n];
    mi = id >> 6; mj = id & 63;
  }
  int packed = ((mi + 4) << 16) | (mj + 4);   // biased positions, packed

  // broadcast the 8 row positions this lane needs (C layout: vgpr v -> M=v+8*hi)
  unsigned riB[8], rjB[8];
#pragma unroll
  for (int v = 0; v < 8; ++v) {
    int pk = __shfl(packed, v + 8 * hi, 32);
    riB[v] = (unsigned)(pk >> 16);
    rjB[v] = (unsigned)(pk & 0xffff);
  }

  // A fragments for all 12 K-steps (lane: M=lo, vgpr v: K = k0 + 2*hi + v)
  const float* rdesc = rowDesc + ((size_t)b * NHW + rowPix) * NCH;
  v2f a[12];
#pragma unroll
  for (int k = 0; k < 12; ++k)
    a[k] = *(const v2f*)(rdesc + k * 4 + 2 * hi);

  float rmax[8];
#pragma unroll
  for (int v = 0; v < 8; ++v) rmax[v] = -1e30f;

  const float* cbase = colDesc + (size_t)b * NHW * NCH;

  for (int t = 0; t < NHW / 16; ++t) {
    int col = t * 16 + lo;
    const float* cd = cbase + (size_t)col * NCH;
    v2f bf[12];
#pragma unroll
    for (int k = 0; k < 12; ++k)
      bf[k] = *(const v2f*)(cd + k * 4 + 2 * hi);
    v8f acc = {0.f, 0.f, 0.f, 0.f, 0.f, 0.f, 0.f, 0.f};
#pragma unroll
    for (int k = 0; k < 12; ++k)
      acc = __builtin_amdgcn_wmma_f32_16x16x4_f32(
          false, a[k], false, bf[k], (short)0, acc, false, false);
    unsigned ci = (unsigned)(col >> 6), cj = (unsigned)(col & 63);
#pragma unroll
    for (int v = 0; v < 8; ++v) {
      unsigned du = riB[v] - ci;                  // wraps huge if negative
      unsigned dj = rjB[v] - cj;
      unsigned mx = du > dj ? du : dj;            // v_max_u32
      float sc = (mx <= 8u) ? -5.0f : acc[v];     // cndmask: exclude window
      rmax[v] = fmaxf(rmax[v], sc);
    }
  }

  // max-reduce across the 16 lanes of each half (rows live per-half)
#pragma unroll
  for (int m = 1; m < 16; m <<= 1) {
#pragma unroll
    for (int v = 0; v < 8; ++v)
      rmax[v] = fmaxf(rmax[v], __shfl_xor(rmax[v], m, 32));
  }
  if (lo == 0) {                  // lanes 0 and 16 write rows 0-7 / 8-15
    float* o = out + (size_t)b * NCORR + rowBase + 8 * hi;
#pragma unroll
    for (int v = 0; v < 8; ++v) o[v] = 2.0f - 2.0f * rmax[v];
  }
}

// ---------------------------------------------------------------------------
// Kernel 3: positive distances + weighted margin loss per image.
// grid: NB blocks of 256 threads.
// ---------------------------------------------------------------------------
__global__ __launch_bounds__(256) void k_loss(
    const float* __restrict__ desc1, const float* __restrict__ desc2,
    const float* __restrict__ att1, const float* __restrict__ att2,
    const int* __restrict__ ids, const int* __restrict__ fp2,
    const float* __restrict__ neg1, const float* __restrict__ neg2,
    float* __restrict__ ratios) {
  int b = blockIdx.x, tid = threadIdx.x;
  __shared__ float sL[256], sW[256];
  const int* idsB = ids + (size_t)b * NCORR;
  const int* fpI  = fp2 + (size_t)b * 2 * NCORR;
  const int* fpJ  = fpI + NCORR;
  float lsum = 0.f, wsum = 0.f;
  for (int n = tid; n < NCORR; n += 256) {
    int id = idsB[n];
    int p2 = fpI[n] * NW + fpJ[n];
    const float4* d1 = (const float4*)(desc1 + ((size_t)b * NHW + id) * NCH);
    const float4* d2 = (const float4*)(desc2 + ((size_t)b * NHW + p2) * NCH);
    float dot = 0.f;
#pragma unroll
    for (int c = 0; c < NCH / 4; ++c) {
      float4 u = d1[c], w4 = d2[c];
      dot += u.x * w4.x + u.y * w4.y + u.z * w4.z + u.w * w4.w;
    }
    float pos = 2.0f - 2.0f * dot;
    float neg = fminf(neg1[(size_t)b * NCORR + n], neg2[(size_t)b * NCORR + n]);
    float w = att1[(size_t)b * NHW + id] * att2[(size_t)b * NHW + p2];
    lsum += w * fmaxf(0.0f, 1.0f + pos - neg);
    wsum += w;
  }
  sL[tid] = lsum; sW[tid] = wsum;
  __syncthreads();
  for (int s = 128; s > 0; s >>= 1) {
    if (tid < s) { sL[tid] += sL[tid + s]; sW[tid] += sW[tid + s]; }
    __syncthreads();
  }
  if (tid == 0) ratios[b] = sL[0] / sW[0];
}

// Kernel 4: deterministic final mean (avoid float atomics).
__global__ void k_mean(const float* __restrict__ ratios, float* __restrict__ out) {
  if (threadIdx.x == 0) {
    float s = 0.f;
    for (int i = 0; i < NB; ++i) s += ratios[i];
    out[0] = s * (1.0f / NB);
  }
}

extern "C" void kernel_launch(void* const* d_in, const int* in_sizes, int n_in,
                              void* d_out, int out_size, void* d_ws, size_t ws_size,
                              hipStream_t stream) {
  (void)in_sizes; (void)n_in; (void)out_size; (void)ws_size;
  const float* x1   = (const float*)d_in[0];
  const float* x2   = (const float*)d_in[1];
  const float* att1 = (const float*)d_in[2];
  const float* att2 = (const float*)d_in[3];
  const int*   ids  = (const int*)d_in[4];
  const int*   fp2  = (const int*)d_in[5];

  float* ws    = (float*)d_ws;
  float* desc1 = ws;                                  // NB*NHW*NCH
  float* desc2 = desc1 + (size_t)NB * NHW * NCH;      // NB*NHW*NCH
  float* neg1  = desc2 + (size_t)NB * NHW * NCH;      // NB*NCORR
  float* neg2  = neg1 + (size_t)NB * NCORR;           // NB*NCORR
  float* ratios = neg2 + (size_t)NB * NCORR;          // NB

  k_normalize<<<dim3(NB * NHW / 256, 2), 256, 0, stream>>>(x1, x2, desc1, desc2);
  k_negmin<<<dim3(NCORR / 64, NB), 128, 0, stream>>>(desc1, desc2, ids, fp2, neg2, 0);
  k_negmin<<<dim3(NCORR / 64, NB), 128, 0, stream>>>(desc2, desc1, ids, fp2, neg1, 1);
  k_loss<<<NB, 256, 0, stream>>>(desc1, desc2, att1, att2, ids, fp2, neg1, neg2, ratios);
  k_mean<<<1, 32, 0, stream>>>(ratios, (float*)d_out);
}